// TransformerLayer_12438225289773
// MI455X (gfx1250) — compile-verified
//
#include <hip/hip_runtime.h>
#include <math.h>

// CDNA5 / gfx1250, wave32. fp32 transformer layer on V_WMMA_F32_16X16X4_F32
// with double-buffered async global->LDS staging (ASYNCcnt).

typedef __attribute__((ext_vector_type(2))) float v2f;
typedef __attribute__((ext_vector_type(8))) float v8f;

#define S_TOK 4096
#define DM    2048
#define DI    512

#define USE_ASYNC_LDS 1

// 32-bit LDS byte offset of a __shared__ object (addrspacecast generic->AS3).
__device__ __forceinline__ unsigned lds_off_u32(const void* p) {
    return (unsigned)(unsigned long long)(const __attribute__((address_space(3))) char*)p;
}

// Per-lane async copy of 16B global -> LDS, tracked by ASYNCcnt.
__device__ __forceinline__ void async_copy_b128(unsigned lds_byte, const float* gptr) {
    unsigned long long ga = (unsigned long long)gptr;
    asm volatile("global_load_async_to_lds_b128 %0, %1, off"
                 :: "v"(lds_byte), "v"(ga) : "memory");
}

__device__ __forceinline__ void wait_async_all() {
    asm volatile("s_wait_asynccnt 0x0" ::: "memory");
}

// ---------------------------------------------------------------------------
// Tiled fp32 WMMA GEMM:  out = act( alpha * (A @ B(T)) + bias ) + residual
//   A: [M,K] row-major
//   BT=true : B is W[N,K] row-major, compute A @ W^T  (stored UNtransposed in
//             LDS; B fragments read it identically to A fragments)
//   BT=false: B is     [K,N] row-major, compute A @ B (attn @ v)
// Block tile 128x128, BK=16, 256 threads (8 waves, 4x2), wave tile 32x64.
// Dims must be multiples of 128 (M,N) / 16 (K) -- true for this layer.
// LDS strides: 20 (A / W tiles, 16B-aligned rows, conflict-free col reads),
//              132 (B[K,N] tiles, 16B-aligned rows).
// ---------------------------------------------------------------------------
template <bool BT>
__global__ __launch_bounds__(256) void gemm_f32_wmma(
    const float* __restrict__ A, const float* __restrict__ B,
    const float* __restrict__ bias, const float* __restrict__ residual,
    float* __restrict__ out, int M, int N, int K, float alpha, int fuse_relu)
{
    const int tid   = threadIdx.x;
    const int lane  = tid & 31;
    const int wave  = tid >> 5;
    const int waveM = wave >> 1;   // 0..3 : 32-row strip
    const int waveN = wave & 1;    // 0..1 : 64-col strip
    const int half  = lane >> 4;   // K-pair select (frags) / +8 rows (C)
    const int lc    = lane & 15;

    const int mBase = blockIdx.y * 128;
    const int nBase = blockIdx.x * 128;

    // double buffers: 2560 floats per buffer per array
    __shared__ float As[2 * 2560];   // 128 x 16, stride 20
    __shared__ float Bs[2 * 2560];   // BT: 128 x 16 stride 20 ; !BT: 16 x 128 stride 132

    v8f acc[2][4];
#pragma unroll
    for (int i = 0; i < 2; ++i)
#pragma unroll
        for (int j = 0; j < 4; ++j)
#pragma unroll
            for (int e = 0; e < 8; ++e) acc[i][j][e] = 0.0f;

    const unsigned asBase = lds_off_u32(As);
    const unsigned bsBase = lds_off_u32(Bs);
    const int nch = K >> 4;

    // issue async staging of one 16-wide K chunk into buffer half `buf`
    auto stage = [&](int k0, int buf) {
#pragma unroll
        for (int i = 0; i < 2; ++i) {                   // A tile 128x16
            int idx = tid + i * 256;                    // 0..511
            int row = idx >> 2;
            int c4  = (idx & 3) << 2;
            async_copy_b128(asBase + (unsigned)((buf * 2560 + row * 20 + c4) * 4),
                            A + (size_t)(mBase + row) * K + k0 + c4);
        }
        if (BT) {                                       // W tile 128x16 (row-major)
#pragma unroll
            for (int i = 0; i < 2; ++i) {
                int idx = tid + i * 256;
                int n   = idx >> 2;
                int c4  = (idx & 3) << 2;
                async_copy_b128(bsBase + (unsigned)((buf * 2560 + n * 20 + c4) * 4),
                                B + (size_t)(nBase + n) * K + k0 + c4);
            }
        } else {                                        // B tile 16x128 (row-major)
#pragma unroll
            for (int i = 0; i < 2; ++i) {
                int idx = tid + i * 256;
                int kk  = idx >> 5;
                int n4  = (idx & 31) << 2;
                async_copy_b128(bsBase + (unsigned)((buf * 2560 + kk * 132 + n4) * 4),
                                B + (size_t)(k0 + kk) * N + nBase + n4);
            }
        }
    };

#if USE_ASYNC_LDS
    stage(0, 0);
    int cur = 0;
    for (int ch = 0; ch < nch; ++ch) {
        wait_async_all();        // our copies of chunk `ch` landed in LDS
        __syncthreads();         // everyone's copies landed; prev reads done
        if (ch + 1 < nch) stage((ch + 1) * 16, cur ^ 1);

        const float* Ab = As + cur * 2560;
        const float* Bb = Bs + cur * 2560;
#pragma unroll
        for (int ks = 0; ks < 4; ++ks) {
            const int kk = ks * 4 + half * 2;
            v2f a[2], b[4];
#pragma unroll
            for (int tm = 0; tm < 2; ++tm) {
                int r = waveM * 32 + tm * 16 + lc;      // lane = M row
                a[tm].x = Ab[r * 20 + kk];
                a[tm].y = Ab[r * 20 + kk + 1];
            }
#pragma unroll
            for (int tn = 0; tn < 4; ++tn) {
                int c = waveN * 64 + tn * 16 + lc;      // lane = N col
                if (BT) {
                    b[tn].x = Bb[c * 20 + kk];
                    b[tn].y = Bb[c * 20 + kk + 1];
                } else {
                    b[tn].x = Bb[(kk + 0) * 132 + c];
                    b[tn].y = Bb[(kk + 1) * 132 + c];
                }
            }
#pragma unroll
            for (int tm = 0; tm < 2; ++tm)
#pragma unroll
                for (int tn = 0; tn < 4; ++tn)
                    acc[tm][tn] = __builtin_amdgcn_wmma_f32_16x16x4_f32(
                        false, a[tm], false, b[tn], (short)0, acc[tm][tn],
                        false, false);
        }
        cur ^= 1;
    }
#else
    // synchronous fallback: VGPR staging, single buffer, two barriers/chunk
    for (int k0 = 0; k0 < K; k0 += 16) {
        float4 ra[2], rb[2];
#pragma unroll
        for (int i = 0; i < 2; ++i) {
            int idx = tid + i * 256;
            int row = idx >> 2, c4 = (idx & 3) << 2;
            ra[i] = *(const float4*)(A + (size_t)(mBase + row) * K + k0 + c4);
            if (BT) {
                rb[i] = *(const float4*)(B + (size_t)(nBase + row) * K + k0 + c4);
            } else {
                int kk = idx >> 5, n4 = (idx & 31) << 2;
                rb[i] = *(const float4*)(B + (size_t)(k0 + kk) * N + nBase + n4);
            }
        }
        __syncthreads();
#pragma unroll
        for (int i = 0; i < 2; ++i) {
            int idx = tid + i * 256;
            int row = idx >> 2, c4 = (idx & 3) << 2;
            As[row * 20 + c4 + 0] = ra[i].x; As[row * 20 + c4 + 1] = ra[i].y;
            As[row * 20 + c4 + 2] = ra[i].z; As[row * 20 + c4 + 3] = ra[i].w;
            if (BT) {
                Bs[row * 20 + c4 + 0] = rb[i].x; Bs[row * 20 + c4 + 1] = rb[i].y;
                Bs[row * 20 + c4 + 2] = rb[i].z; Bs[row * 20 + c4 + 3] = rb[i].w;
            } else {
                int kk = idx >> 5, n4 = (idx & 31) << 2;
                Bs[kk * 132 + n4 + 0] = rb[i].x; Bs[kk * 132 + n4 + 1] = rb[i].y;
                Bs[kk * 132 + n4 + 2] = rb[i].z; Bs[kk * 132 + n4 + 3] = rb[i].w;
            }
        }
        __syncthreads();
        if (k0 + 16 < K)
            __builtin_prefetch(A + (size_t)(mBase + (tid >> 1)) * K + k0 + 16, 0, 1);
#pragma unroll
        for (int ks = 0; ks < 4; ++ks) {
            const int kk = ks * 4 + half * 2;
            v2f a[2], b[4];
#pragma unroll
            for (int tm = 0; tm < 2; ++tm) {
                int r = waveM * 32 + tm * 16 + lc;
                a[tm].x = As[r * 20 + kk];
                a[tm].y = As[r * 20 + kk + 1];
            }
#pragma unroll
            for (int tn = 0; tn < 4; ++tn) {
                int c = waveN * 64 + tn * 16 + lc;
                if (BT) { b[tn].x = Bs[c * 20 + kk];  b[tn].y = Bs[c * 20 + kk + 1]; }
                else    { b[tn].x = Bs[kk * 132 + c]; b[tn].y = Bs[(kk + 1) * 132 + c]; }
            }
#pragma unroll
            for (int tm = 0; tm < 2; ++tm)
#pragma unroll
                for (int tn = 0; tn < 4; ++tn)
                    acc[tm][tn] = __builtin_amdgcn_wmma_f32_16x16x4_f32(
                        false, a[tm], false, b[tn], (short)0, acc[tm][tn],
                        false, false);
        }
    }
#endif

    // ---- epilogue: alpha, bias, relu, residual ----
    // C/D layout: VGPR i -> M=i (lanes 0-15) / M=i+8 (lanes 16-31), N=lane&15
#pragma unroll
    for (int tm = 0; tm < 2; ++tm) {
#pragma unroll
        for (int tn = 0; tn < 4; ++tn) {
#pragma unroll
            for (int i = 0; i < 8; ++i) {
                int r = mBase + waveM * 32 + tm * 16 + half * 8 + i;
                int c = nBase + waveN * 64 + tn * 16 + lc;
                float v = alpha * acc[tm][tn][i];
                if (bias)      v += bias[c];
                if (fuse_relu) v = fmaxf(v, 0.0f);
                if (residual)  v += residual[(size_t)r * N + c];
                out[(size_t)r * N + c] = v;
            }
        }
    }
}

// ---------------------------------------------------------------------------
// In-place row softmax; one block per row (scale already applied by GEMM).
// ---------------------------------------------------------------------------
__global__ __launch_bounds__(256) void softmax_rows(float* __restrict__ s, int ncols)
{
    __shared__ float red[256];
    const int tid = threadIdx.x;
    float* row = s + (size_t)blockIdx.x * ncols;

    float m = -INFINITY;
    for (int c = tid; c < ncols; c += 256) m = fmaxf(m, row[c]);
    red[tid] = m;
    __syncthreads();
    for (int off = 128; off > 0; off >>= 1) {
        if (tid < off) red[tid] = fmaxf(red[tid], red[tid + off]);
        __syncthreads();
    }
    m = red[0];
    __syncthreads();

    float sum = 0.0f;
    for (int c = tid; c < ncols; c += 256) {
        float e = __expf(row[c] - m);
        row[c] = e;
        sum += e;
    }
    red[tid] = sum;
    __syncthreads();
    for (int off = 128; off > 0; off >>= 1) {
        if (tid < off) red[tid] += red[tid + off];
        __syncthreads();
    }
    const float inv = 1.0f / red[0];
    for (int c = tid; c < ncols; c += 256) row[c] *= inv;
}

// ---------------------------------------------------------------------------
extern "C" void kernel_launch(void* const* d_in, const int* in_sizes, int n_in,
                              void* d_out, int out_size, void* d_ws, size_t ws_size,
                              hipStream_t stream)
{
    (void)in_sizes; (void)n_in; (void)out_size; (void)ws_size;

    const float* x  = (const float*)d_in[0];
    const float* qw = (const float*)d_in[1];
    const float* qb = (const float*)d_in[2];
    const float* kw = (const float*)d_in[3];
    const float* kb = (const float*)d_in[4];
    const float* vw = (const float*)d_in[5];
    const float* vb = (const float*)d_in[6];
    const float* w1 = (const float*)d_in[7];
    const float* b1 = (const float*)d_in[8];
    const float* w2 = (const float*)d_in[9];
    const float* b2 = (const float*)d_in[10];

    // output = hidden [S,DM] ++ k [S,DI] ++ v [S,DM]
    float* hidden = (float*)d_out;
    float* kout   = hidden + (size_t)S_TOK * DM;
    float* vout   = kout + (size_t)S_TOK * DI;

    // workspace layout
    char*  ws     = (char*)d_ws;
    float* q_ws   = (float*)ws;                                   //  8 MiB
    float* s_ws   = (float*)(ws + (size_t)S_TOK * DI * 4);        // 64 MiB (L2-resident)
    float* res_ws = (float*)(ws + (size_t)S_TOK * DI * 4
                                + (size_t)S_TOK * S_TOK * 4);     // 32 MiB
    float* t_ws   = (float*)(ws + (size_t)S_TOK * DI * 4
                                + (size_t)S_TOK * S_TOK * 4
                                + (size_t)S_TOK * DM * 4);        // 32 MiB

    const dim3 blk(256);
    const float scale = 0.044194173824159216f;  // 1/sqrt(512)

    // q,k,v projections (k and v written straight into d_out slices)
    gemm_f32_wmma<true><<<dim3(DI / 128, S_TOK / 128), blk, 0, stream>>>(
        x, qw, qb, nullptr, q_ws, S_TOK, DI, DM, 1.0f, 0);
    gemm_f32_wmma<true><<<dim3(DI / 128, S_TOK / 128), blk, 0, stream>>>(
        x, kw, kb, nullptr, kout, S_TOK, DI, DM, 1.0f, 0);
    gemm_f32_wmma<true><<<dim3(DM / 128, S_TOK / 128), blk, 0, stream>>>(
        x, vw, vb, nullptr, vout, S_TOK, DM, DM, 1.0f, 0);

    // scores = (q @ k^T) * scale   (k acts as the [N,K] "weight")
    gemm_f32_wmma<true><<<dim3(S_TOK / 128, S_TOK / 128), blk, 0, stream>>>(
        q_ws, kout, nullptr, nullptr, s_ws, S_TOK, S_TOK, DI, scale, 0);

    softmax_rows<<<dim3(S_TOK), blk, 0, stream>>>(s_ws, S_TOK);

    // x_residual = P @ v + x      (v is [K,N] row-major, non-transposed)
    gemm_f32_wmma<false><<<dim3(DM / 128, S_TOK / 128), blk, 0, stream>>>(
        s_ws, vout, nullptr, x, res_ws, S_TOK, DM, S_TOK, 1.0f, 0);

    // t = relu(x_residual @ w1^T + b1)
    gemm_f32_wmma<true><<<dim3(DM / 128, S_TOK / 128), blk, 0, stream>>>(
        res_ws, w1, b1, nullptr, t_ws, S_TOK, DM, DM, 1.0f, 1);

    // hidden = t @ w2^T + b2 + x_residual
    gemm_f32_wmma<true><<<dim3(DM / 128, S_TOK / 128), blk, 0, stream>>>(
        t_ws, w2, b2, res_ws, hidden, S_TOK, DM, DM, 1.0f, 0);
}